// GaussianSoftVoxelization_83537113907647
// MI455X (gfx1250) — compile-verified
//
#include <hip/hip_runtime.h>
#include <stdint.h>

#define GX 432
#define GY 496
#define NCELL (GX * GY)      // 214272
#define MAXV 20000
#define MAXP 35
#define WCAP 128             // per-wave LDS key capacity (Poisson(~7) tail << 128)
#define CHUNK 1024           // cells per scan block (256 thr x 4)
#define NCHUNK ((NCELL + CHUNK - 1) / CHUNK)   // 210

typedef unsigned long long u64;

// ---------------------------------------------------------------- init
__global__ void gsv_init(float* __restrict__ out, int outN,
                         unsigned* __restrict__ gridCount,
                         unsigned* __restrict__ cursor,
                         int* __restrict__ cellList) {
    int total = outN > NCELL ? outN : NCELL;
    for (int i = blockIdx.x * blockDim.x + threadIdx.x; i < total;
         i += gridDim.x * blockDim.x) {
        if (i < outN) out[i] = 0.0f;
        if (i < NCELL) { gridCount[i] = 0u; cursor[i] = 0u; }
        if (i < MAXV)  cellList[i] = -1;
    }
}

// ---------------------------------------------------------------- pass 1: bin + weight
__global__ void gsv_p1(const float4* __restrict__ pts,
                       const float* __restrict__ sigmap, int N,
                       unsigned* __restrict__ gridCount,
                       int* __restrict__ plin, float* __restrict__ pw) {
    const float sig = fmaxf(sigmap[0], 0.001f);
    for (int i = blockIdx.x * blockDim.x + threadIdx.x; i < N;
         i += gridDim.x * blockDim.x) {
        __builtin_prefetch(pts + i + 4096, 0, 0);   // global_prefetch_b8
        float4 p = pts[i];
        float fx = floorf((p.x - 0.0f)   / 0.16f);
        float fy = floorf((p.y + 39.68f) / 0.16f);
        float fz = floorf((p.z + 3.0f)   / 4.0f);
        int vx = (int)fx, vy = (int)fy, vz = (int)fz;
        bool valid = (vx >= 0) && (vx < GX) && (vy >= 0) && (vy < GY) &&
                     (vz >= 0) && (vz < 1);
        if (valid) {
            int lin = vx * GY + vy;
            atomicAdd(&gridCount[lin], 1u);
            float cx = (fx + 0.5f) * 0.16f + 0.0f;
            float cy = (fy + 0.5f) * 0.16f - 39.68f;
            float cz = (fz + 0.5f) * 4.0f - 3.0f;
            float dx = p.x - cx, dy = p.y - cy, dz = p.z - cz;
            float dist = sqrtf(dx * dx + dy * dy + dz * dz);
            float t = dist / sig;
            plin[i] = lin;
            pw[i] = expf(-0.5f * t * t);
        } else {
            plin[i] = -1;
        }
    }
}

// ---------------------------------------------------------------- scan phase 1: chunk totals
__global__ void gsv_scan1(const unsigned* __restrict__ gridCount,
                          unsigned* __restrict__ partF,
                          unsigned* __restrict__ partC) {
    __shared__ unsigned aF[256], aC[256];
    const int t = threadIdx.x, b = blockIdx.x;
    const int base = b * CHUNK + t * 4;
    unsigned tf = 0u, tc = 0u;
    #pragma unroll
    for (int e = 0; e < 4; ++e) {
        int i = base + e;
        unsigned c = (i < NCELL) ? gridCount[i] : 0u;
        tc += c; tf += (c ? 1u : 0u);
    }
    aF[t] = tf; aC[t] = tc;
    __syncthreads();
    for (int ofs = 128; ofs > 0; ofs >>= 1) {
        if (t < ofs) { aF[t] += aF[t + ofs]; aC[t] += aC[t + ofs]; }
        __syncthreads();
    }
    if (t == 0) { partF[b] = aF[0]; partC[b] = aC[0]; }
}

// ---------------------------------------------------------------- scan phase 2: scan 210 partials
__global__ void gsv_scan2(unsigned* __restrict__ partF,
                          unsigned* __restrict__ partC,
                          unsigned* __restrict__ scanF,
                          unsigned* __restrict__ scanC) {
    if (threadIdx.x == 0) {
        unsigned rf = 0u, rc = 0u;
        for (int b = 0; b < NCHUNK; ++b) {
            scanF[b] = rf; scanC[b] = rc;
            rf += partF[b]; rc += partC[b];
        }
    }
}

// ---------------------------------------------------------------- scan phase 3: local scan + compact
__global__ void gsv_scan3(const unsigned* __restrict__ gridCount,
                          const unsigned* __restrict__ scanF,
                          const unsigned* __restrict__ scanC,
                          unsigned* __restrict__ gridRank,
                          unsigned* __restrict__ gridOff,
                          int* __restrict__ cellList) {
    __shared__ unsigned aF[256], aC[256];
    const int t = threadIdx.x, b = blockIdx.x;
    const int base = b * CHUNK + t * 4;
    unsigned f[4], c[4], tf = 0u, tc = 0u;
    #pragma unroll
    for (int e = 0; e < 4; ++e) {
        int i = base + e;
        c[e] = (i < NCELL) ? gridCount[i] : 0u;
        f[e] = c[e] ? 1u : 0u;
        tc += c[e]; tf += f[e];
    }
    aF[t] = tf; aC[t] = tc;
    __syncthreads();
    for (int ofs = 1; ofs < 256; ofs <<= 1) {
        unsigned x = (t >= ofs) ? aF[t - ofs] : 0u;
        unsigned y = (t >= ofs) ? aC[t - ofs] : 0u;
        __syncthreads();
        aF[t] += x; aC[t] += y;
        __syncthreads();
    }
    unsigned exF = aF[t] - tf + scanF[b];   // exclusive prefix for first element
    unsigned exC = aC[t] - tc + scanC[b];
    #pragma unroll
    for (int e = 0; e < 4; ++e) {
        int i = base + e;
        if (i < NCELL) {
            gridRank[i] = exF;
            gridOff[i]  = exC;
            if (f[e] && exF < MAXV) cellList[exF] = i;   // compact kept voxels
        }
        exF += f[e]; exC += c[e];
    }
}

// ---------------------------------------------------------------- pass 2: CSR key scatter
// key = { order-mapped w bits , ~pid } : max key == (max w, then min original index)
__global__ void gsv_p2(int N, const int* __restrict__ plin,
                       const float* __restrict__ pw,
                       const unsigned* __restrict__ gridOff,
                       unsigned* __restrict__ cursor,
                       u64* __restrict__ ptKey) {
    for (int i = blockIdx.x * blockDim.x + threadIdx.x; i < N;
         i += gridDim.x * blockDim.x) {
        int lin = plin[i];
        if (lin >= 0) {
            unsigned slot = atomicAdd(&cursor[lin], 1u);
            unsigned o = gridOff[lin] + slot;
            unsigned hi = __float_as_uint(pw[i]) ^ 0x80000000u;  // w>=0 -> hi>=2^31 (never 0)
            unsigned lo = 0xFFFFFFFFu - (unsigned)i;
            ptKey[o] = ((u64)hi << 32) | lo;
        }
    }
}

// ---------------------------------------------------------------- pass V: wave-per-kept-voxel top-k
__global__ void gsv_vox(const float4* __restrict__ pts,
                        const unsigned* __restrict__ gridCount,
                        const unsigned* __restrict__ gridOff,
                        const int* __restrict__ cellList,
                        u64* __restrict__ ptKey,
                        float* __restrict__ feats,
                        int* __restrict__ coords,
                        int* __restrict__ counts) {
    __shared__ u64      skey[8 * WCAP];
    __shared__ unsigned sPid[8 * MAXP];
    __shared__ float    sW[8 * MAXP];
    __shared__ float    sSum[8];

    const int wave = threadIdx.x >> 5;
    const int lane = threadIdx.x & 31;
    const int rank = blockIdx.x * 8 + wave;      // exactly MAXV waves launched
    if (rank >= MAXV) return;
    const int cell = cellList[rank];
    if (cell < 0) return;                        // fewer than MAXV occupied voxels

    const unsigned cnt = gridCount[cell];
    const unsigned off = gridOff[cell];
    const int k = (int)(cnt < MAXP ? cnt : MAXP);
    const bool inlds = (cnt <= WCAP);

    volatile u64* kvL = &skey[wave * WCAP];
    volatile u64* kvG = (volatile u64*)(ptKey + off);
    volatile unsigned* vPid = &sPid[wave * MAXP];
    volatile float*    vW   = &sW[wave * MAXP];

    if (inlds) {
        // CDNA5 async global->LDS gather of this voxel's (w,pid) keys
        for (int j = lane; j < (int)cnt; j += 32) {
            uint32_t ldsoff = (uint32_t)(uintptr_t)&skey[wave * WCAP + j];
            uint64_t ga = (uint64_t)(uintptr_t)(ptKey + off + (unsigned)j);
            asm volatile("global_load_async_to_lds_b64 %0, %1, off"
                         :: "v"(ldsoff), "v"(ga) : "memory");
        }
        asm volatile("s_wait_asynccnt 0" ::: "memory");
    }

    // cooperative top-k: rank by (w desc, pid asc) -- matches stable lexsort((-w, seg))
    float sum = 0.0f;
    for (int r = 0; r < k; ++r) {
        unsigned lHi = 0u, lLo = 0u; int lJ = -1;
        for (int j = lane; j < (int)cnt; j += 32) {
            u64 key = inlds ? kvL[j] : kvG[j];
            unsigned hi = (unsigned)(key >> 32), lo = (unsigned)key;
            if (hi | lo) {
                if (hi > lHi || (hi == lHi && lo > lLo)) { lHi = hi; lLo = lo; lJ = j; }
            }
        }
        unsigned gHi = lHi, gLo = lLo;
        for (int d = 16; d > 0; d >>= 1) {
            unsigned ohi = __shfl_xor(gHi, d, 32);
            unsigned olo = __shfl_xor(gLo, d, 32);
            if (ohi > gHi || (ohi == gHi && olo > gLo)) { gHi = ohi; gLo = olo; }
        }
        // unique owner lane (keys are unique) retires the winner
        if (lJ >= 0 && lHi == gHi && lLo == gLo) {
            if (inlds) kvL[lJ] = 0ull; else kvG[lJ] = 0ull;
        }
        float wbest = __uint_as_float(gHi ^ 0x80000000u);
        if (lane == 0) {
            vPid[r] = 0xFFFFFFFFu - gLo;   // original point index
            vW[r]   = wbest;
            sum += wbest;
        }
        __builtin_amdgcn_wave_barrier();
    }

    if (lane == 0) sSum[wave] = sum;
    __builtin_amdgcn_wave_barrier();
    const float denom = ((volatile float*)sSum)[wave] + 1e-6f;

    // feats[rank, r, 0:4] = point * wn  (one B128 store per kept slot)
    for (int r = lane; r < k; r += 32) {
        unsigned pid = vPid[r];
        float wn = vW[r] / denom;
        float4 p = pts[pid];
        float4 o; o.x = p.x * wn; o.y = p.y * wn; o.z = p.z * wn; o.w = p.w * wn;
        ((float4*)feats)[(size_t)rank * MAXP + (size_t)r] = o;
    }
    if (lane == 0) {
        coords[rank * 3 + 0] = cell / GY;
        coords[rank * 3 + 1] = cell % GY;
        coords[rank * 3 + 2] = 0;
        counts[rank] = k;
    }
}

// ---------------------------------------------------------------- launch
extern "C" void kernel_launch(void* const* d_in, const int* in_sizes, int n_in,
                              void* d_out, int out_size, void* d_ws, size_t ws_size,
                              hipStream_t stream) {
    const float4* pts   = (const float4*)d_in[0];
    const float*  sigma = (const float*)d_in[1];
    const int N = in_sizes[0] / 4;

    float* out    = (float*)d_out;
    float* feats  = out;                                   // [MAXV, MAXP, 4] f32
    int*   coords = (int*)(out + (size_t)MAXV * MAXP * 4); // [MAXV, 3] i32
    int*   counts = coords + (size_t)MAXV * 3;             // [MAXV] i32

    // workspace carve (8-byte alignment preserved: every region is a multiple of 8B)
    unsigned* gridCount = (unsigned*)d_ws;                 // NCELL
    unsigned* cursor    = gridCount + NCELL;               // NCELL
    unsigned* gridRank  = cursor + NCELL;                  // NCELL
    unsigned* gridOff   = gridRank + NCELL;                // NCELL
    unsigned* partF     = gridOff + NCELL;                 // 256 (NCHUNK used)
    unsigned* partC     = partF + 256;                     // 256
    unsigned* scanF     = partC + 256;                     // 256
    unsigned* scanC     = scanF + 256;                     // 256
    int*      cellList  = (int*)(scanC + 256);             // MAXV
    u64*      ptKey     = (u64*)(cellList + MAXV);         // N  (MAXV=20000 -> 8B aligned)
    int*      plin      = (int*)(ptKey + N);               // N
    float*    pw        = (float*)(plin + N);              // N

    gsv_init<<<2048, 256, 0, stream>>>(out, out_size, gridCount, cursor, cellList);
    gsv_p1<<<4096, 256, 0, stream>>>(pts, sigma, N, gridCount, plin, pw);
    gsv_scan1<<<NCHUNK, 256, 0, stream>>>(gridCount, partF, partC);
    gsv_scan2<<<1, 64, 0, stream>>>(partF, partC, scanF, scanC);
    gsv_scan3<<<NCHUNK, 256, 0, stream>>>(gridCount, scanF, scanC, gridRank, gridOff, cellList);
    gsv_p2<<<4096, 256, 0, stream>>>(N, plin, pw, gridOff, cursor, ptKey);
    gsv_vox<<<(MAXV + 7) / 8, 256, 0, stream>>>(pts, gridCount, gridOff, cellList,
                                                ptKey, feats, coords, counts);
}